// SelfDistill_53841710023017
// MI455X (gfx1250) — compile-verified
//
#include <hip/hip_runtime.h>
#include <math.h>

#define NB   32
#define NC   64
#define NH   128
#define NW   192
#define HM   256
#define WM   384
#define NP   16      // pairs
#define DSC  14.0f
#define EPSC 1e-5f

typedef __attribute__((ext_vector_type(2))) float v2f;
typedef __attribute__((ext_vector_type(8))) float v8f;

// ds_swizzle group-of-32 pattern SWAPX16: and=0x1F, or=0x00, xor=0x10
#define SWIZ_SWAPX16 0x401F

__global__ void sd_init_ws(float* ws) {
    int i = threadIdx.x;
    if (i < 2 * NP) ws[i] = 0.0f;
}

// Block: 256 threads = 8 wave32; wave w handles channels [8w, 8w+8),
// lanes handle 32 consecutive x pixels of row y for pair p.
__global__ __launch_bounds__(256) void sd_warp_loss(
    const float* __restrict__ inp,   // (B,C,H,W)
    const float* __restrict__ cam,   // (B,3,3)
    const float* __restrict__ mask,  // (B,1,HM,WM)
    const int*   __restrict__ flips, // (B,)
    float* __restrict__ ws)          // [0..NP): masked sq-diff sum, [NP..2NP): mask count
{
    const int p    = blockIdx.z;
    const int y    = blockIdx.y;
    const int tid  = threadIdx.x;
    const int lane = tid & 31;
    const int wave = tid >> 5;
    const int x    = blockIdx.x * 32 + lane;

    const int b0 = 2 * p, b1 = 2 * p + 1;

    // ---- per-pair scalar warp parameters (uniform -> scalarized) ----
    const float fx0 = cam[b0 * 9 + 0] / DSC;
    const float fx1 = cam[b1 * 9 + 0] / DSC;
    float       cx0 = cam[b0 * 9 + 2] / DSC;
    const float cx1 = cam[b1 * 9 + 2] / DSC;
    const float cy0 = cam[b0 * 9 + 5] / DSC;
    const float cy1 = cam[b1 * 9 + 5] / DSC;
    const bool flip = ((flips[b0] != 0) != (flips[b1] != 0));
    if (flip) cx0 = (float)NW - cx0;

    const float zoom = fx1 / fx0;
    const float inv  = 1.0f / zoom;
    const float zW   = floorf((float)NW * zoom);
    const float zH   = floorf((float)NH * zoom);
    const float chL  = truncf(cx1 - (cx0 - 0.5f) * zoom - 0.5f);
    const float chT  = truncf(cy1 - (cy0 - 0.5f) * zoom - 0.5f);

    // ---- per-pixel sampling coordinates ----
    const float X = (float)x - chL;
    const float Y = (float)y - chT;
    const bool valid = (X >= 0.0f) & (X < zW) & (Y >= 0.0f) & (Y < zH);

    const float sx = fmaxf((X + 0.5f) * inv - 0.5f, 0.0f);
    const float sy = fmaxf((Y + 0.5f) * inv - 0.5f, 0.0f);
    const int x0 = min(max((int)floorf(sx), 0), NW - 1);
    const int y0 = min(max((int)floorf(sy), 0), NH - 1);
    const int x1 = min(x0 + 1, NW - 1);
    const int y1 = min(y0 + 1, NH - 1);
    const float wx = sx - (float)x0;
    const float wy = sy - (float)y0;

    // nearest-warped mask (mbig[b][yy][xx] == mask[b][2yy][2xx])
    const int xn = min(max((int)floorf(X * inv), 0), NW - 1);
    const int yn = min(max((int)floorf(Y * inv), 0), NH - 1);
    const int xs_n = flip ? (NW - 1 - xn) : xn;
    const float m0w = valid
        ? mask[(size_t)b0 * HM * WM + (size_t)(2 * yn) * WM + (size_t)(2 * xs_n)]
        : 0.0f;
    const float m1 = mask[(size_t)b1 * HM * WM + (size_t)(2 * y) * WM + (size_t)(2 * x)];
    const bool pm = (m0w != 0.0f) & (m1 != 0.0f);   // pm implies valid

    const int xs0 = flip ? (NW - 1 - x0) : x0;
    const int xs1 = flip ? (NW - 1 - x1) : x1;
    const float w00 = (1.0f - wy) * (1.0f - wx);
    const float w01 = (1.0f - wy) * wx;
    const float w10 = wy * (1.0f - wx);
    const float w11 = wy * wx;

    const size_t base0 = (size_t)b0 * NC * NH * NW;
    const size_t base1 = (size_t)b1 * NC * NH * NW;
    const int r0 = y0 * NW;
    const int r1 = y1 * NW;

    float acc = 0.0f;
    if (pm) {
        #pragma unroll
        for (int i = 0; i < 8; ++i) {
            const int c = wave * 8 + i;
            const float* p0 = inp + base0 + (size_t)c * (NH * NW);
            const float v00 = p0[r0 + xs0];
            const float v01 = p0[r0 + xs1];
            const float v10 = p0[r1 + xs0];
            const float v11 = p0[r1 + xs1];
            const float bil = w00 * v00 + w01 * v01 + w10 * v10 + w11 * v11;
            const float tv  = inp[base1 + (size_t)c * (NH * NW) + (size_t)y * NW + x];
            const float d   = bil - tv;
            acc = fmaf(d, d, acc);
        }
    }

    // ---- wave32 reduction via V_WMMA_F32_16X16X4_F32 (fp32-exact) ----
    // A (16x4 f32): lane m<16 holds (M=m,K=0)=acc; lane 16+m holds (M=m,K=2)=acc
    //   -> rowsum(M=m) = acc[m] + acc[m+16].
    // B = ones(4x16) -> D[m,n] = rowsum(m) for every column n.
    v2f a;  a.x = acc;  a.y = 0.0f;
    v2f bb; bb.x = 1.0f; bb.y = 1.0f;
    v8f cz = {};
    v8f dm = __builtin_amdgcn_wmma_f32_16x16x4_f32(
        /*neg_a=*/false, a, /*neg_b=*/false, bb,
        /*c_mod=*/(short)0, cz, /*reuse_a=*/false, /*reuse_b=*/false);
    // lanes 0-15 hold rows 0-7 in D VGPRs 0-7; lanes 16-31 hold rows 8-15.
    const float half8 = dm[0] + dm[1] + dm[2] + dm[3] + dm[4] + dm[5] + dm[6] + dm[7];
    // cross-half add: single ds_swizzle SWAPX16 (no index-clamp glue)
    const float other = __int_as_float(
        __builtin_amdgcn_ds_swizzle(__float_as_int(half8), SWIZ_SWAPX16));
    const float wtot = half8 + other;

    __shared__ float s_wave[8];
    __shared__ float s_cnt;
    if (lane == 0) s_wave[wave] = wtot;

    // mask count (depends only on x,y): wave 0 via ballot+popcount
    if (wave == 0) {
        const unsigned long long bm = __ballot(pm);
        if (lane == 0) s_cnt = (float)__popcll(bm);
    }
    __syncthreads();

    if (tid == 0) {
        const float tot = s_wave[0] + s_wave[1] + s_wave[2] + s_wave[3] +
                          s_wave[4] + s_wave[5] + s_wave[6] + s_wave[7];
        atomicAdd(&ws[p], tot);
        atomicAdd(&ws[NP + p], s_cnt);
    }
}

__global__ void sd_finalize(const float* __restrict__ ws, float* __restrict__ out) {
    const int p = threadIdx.x;
    if (p < NP) {
        const float den = fmaxf(ws[NP + p], 1.0f);
        const float v = sqrtf((ws[p] / (float)NC) / den + EPSC);
        out[p]      = v;   // loss(in0, in1)
        out[p + NP] = v;   // loss(in1, in0): identical in forward
    }
}

extern "C" void kernel_launch(void* const* d_in, const int* in_sizes, int n_in,
                              void* d_out, int out_size, void* d_ws, size_t ws_size,
                              hipStream_t stream) {
    const float* inp   = (const float*)d_in[0];
    const float* cam   = (const float*)d_in[1];
    const float* mask  = (const float*)d_in[2];
    const int*   flips = (const int*)d_in[3];
    float* ws  = (float*)d_ws;
    float* out = (float*)d_out;

    sd_init_ws<<<1, 32, 0, stream>>>(ws);
    dim3 grid(NW / 32, NH, NP);
    sd_warp_loss<<<grid, 256, 0, stream>>>(inp, cam, mask, flips, ws);
    sd_finalize<<<1, 32, 0, stream>>>(ws, out);
}